// MultiScaleCLIPEncoder_40638980554957
// MI455X (gfx1250) — compile-verified
//
#include <hip/hip_runtime.h>
#include <math.h>

typedef float v2f __attribute__((ext_vector_type(2)));
typedef float v4f __attribute__((ext_vector_type(4)));
typedef float v8f __attribute__((ext_vector_type(8)));

// ---------------------------------------------------------------------------
// Generic WMMA fp32 GEMM.
//   TRANSB=1 : Y[M,N] = X[M,K] @ W[N,K]^T + bias        (linear layers)
//   TRANSB=0 : Y[M,N] = X[M,K] @ W[K,N]   + bias        (weight fusion A@B)
//   GATHER=1 : X rows/cols gathered from features via block-rearrange map
// Output row remap: orow = (m / rpb_out)*ob_out + oo_out + (m % rpb_out)
// All of M,N divisible by 64 and K divisible by 32 (checked offline).
// ---------------------------------------------------------------------------
template<bool TRANSB, bool GATHER>
__global__ __launch_bounds__(256)
void gemm_wmma(const float* __restrict__ X, const float* __restrict__ W,
               const float* __restrict__ bias, float* __restrict__ Y,
               int M, int N, int K,
               int rpb_out, int ob_out, int oo_out,
               int g_rpb, int g_gw2, int g_sh)
{
    (void)M;
    constexpr int XSR = 36;                 // LDS row stride (floats): 144B = 9*16
    __shared__ float Xs[64 * XSR];
    __shared__ float Ws[64 * XSR];          // always stores W^T tile: Ws[n][k]

    const int tid  = threadIdx.x;
    const int lane = tid & 31;
    const int wave = tid >> 5;              // 0..7
    const int l15  = lane & 15;
    const int hi   = lane >> 4;             // 0 or 1
    const int m0   = blockIdx.y * 64;
    const int n0   = blockIdx.x * 64;
    const int msub = wave & 3;              // 0..3 (M sub-tile)
    const int nsub = (wave >> 2) << 1;      // 0 or 2 (first of two N sub-tiles)

    v8f acc0 = {};
    v8f acc1 = {};

    for (int kc = 0; kc < K; kc += 32) {
        // ---- stage X tile 64x32 (512 float4 loads by 256 threads) ----
        #pragma unroll
        for (int it = 0; it < 2; ++it) {
            int lin = tid + it * 256;       // 0..511
            int row = lin >> 3;             // 0..63
            int c4  = (lin & 7) << 2;       // 0..28
            int m = m0 + row;
            int k = kc + c4;
            size_t addr;
            if constexpr (GATHER) {
                int b   = m / g_rpb;  int r  = m - b * g_rpb;
                int by  = r / g_gw2;  int bx = r - by * g_gw2;
                int blk = k / 768;    int d  = k - blk * 768;
                int iy  = blk / g_sh; int ix = blk - iy * g_sh;
                int srow = b * 576 + (by * g_sh + iy) * 24 + (bx * g_sh + ix);
                addr = (size_t)srow * 768 + d;
            } else {
                addr = (size_t)m * K + k;
            }
            *(v4f*)&Xs[row * XSR + c4] = *(const v4f*)(X + addr);
        }
        // ---- stage W^T tile into Ws[n][k] ----
        if constexpr (TRANSB) {
            #pragma unroll
            for (int it = 0; it < 2; ++it) {
                int lin = tid + it * 256;
                int row = lin >> 3;         // n: 0..63
                int c4  = (lin & 7) << 2;   // k: 0..28
                *(v4f*)&Ws[row * XSR + c4] =
                    *(const v4f*)(W + (size_t)(n0 + row) * K + (kc + c4));
            }
        } else {
            #pragma unroll
            for (int it = 0; it < 2; ++it) {
                int lin  = tid + it * 256;
                int krow = lin >> 4;            // k: 0..31
                int c4   = (lin & 15) << 2;     // n: 0..60
                v4f v = *(const v4f*)(W + (size_t)(kc + krow) * N + (n0 + c4));
                Ws[(c4 + 0) * XSR + krow] = v.x;
                Ws[(c4 + 1) * XSR + krow] = v.y;
                Ws[(c4 + 2) * XSR + krow] = v.z;
                Ws[(c4 + 3) * XSR + krow] = v.w;
            }
        }
        __syncthreads();

        // ---- 8 WMMA K=4 steps over this 32-wide chunk ----
        #pragma unroll
        for (int ks = 0; ks < 8; ++ks) {
            int kk = (ks << 2) + (hi << 1);
            v2f a  = *(const v2f*)&Xs[(msub * 16 + l15) * XSR + kk];
            v2f b0 = *(const v2f*)&Ws[((nsub + 0) * 16 + l15) * XSR + kk];
            v2f b1 = *(const v2f*)&Ws[((nsub + 1) * 16 + l15) * XSR + kk];
            acc0 = __builtin_amdgcn_wmma_f32_16x16x4_f32(
                false, a, false, b0, (short)0, acc0, false, false);
            acc1 = __builtin_amdgcn_wmma_f32_16x16x4_f32(
                false, a, false, b1, (short)0, acc1, false, false);
        }
        __syncthreads();
    }

    // ---- epilogue: C/D layout => VGPR r: lanes0-15 M=r, lanes16-31 M=r+8 ----
    #pragma unroll
    for (int r = 0; r < 8; ++r) {
        int m   = m0 + msub * 16 + r + hi * 8;
        int n_a = n0 + (nsub + 0) * 16 + l15;
        int n_b = n0 + (nsub + 1) * 16 + l15;
        int bb   = m / rpb_out;
        int orow = bb * ob_out + oo_out + (m - bb * rpb_out);
        float va = acc0[r] + (bias ? bias[n_a] : 0.0f);
        float vb = acc1[r] + (bias ? bias[n_b] : 0.0f);
        Y[(size_t)orow * N + n_a] = va;
        Y[(size_t)orow * N + n_b] = vb;
    }
}

// ---------------------------------------------------------------------------
// bf[l][n] = sum_e vp_w[n,e] * p_l_b[e] + vp_b[n]
// ---------------------------------------------------------------------------
__global__ __launch_bounds__(256)
void bias_fuse(const float* __restrict__ vp_w, const float* __restrict__ vp_b,
               const float* __restrict__ b1, const float* __restrict__ b2,
               const float* __restrict__ b3, float* __restrict__ bf)
{
    __shared__ float red[256];
    int n = blockIdx.x, l = blockIdx.y;
    const float* pb = (l == 0) ? b1 : (l == 1) ? b2 : b3;
    float s = 0.0f;
    for (int e = threadIdx.x; e < 1024; e += 256)
        s += vp_w[(size_t)n * 1024 + e] * pb[e];
    red[threadIdx.x] = s;
    __syncthreads();
    for (int st = 128; st; st >>= 1) {
        if (threadIdx.x < st) red[threadIdx.x] += red[threadIdx.x + st];
        __syncthreads();
    }
    if (threadIdx.x == 0) bf[l * 1024 + n] = red[0] + vp_b[n];
}

// ---------------------------------------------------------------------------
// Row LayerNorm over 1024 cols, optional exact GELU. Safe in-place.
// ---------------------------------------------------------------------------
template<bool GELU>
__global__ __launch_bounds__(256)
void ln_kernel(const float* __restrict__ in, float* __restrict__ out,
               const float* __restrict__ g, const float* __restrict__ b)
{
    __shared__ float red[256];
    size_t row = blockIdx.x;
    const float* x = in + row * 1024;
    float v[4];
    float s = 0.0f;
    #pragma unroll
    for (int i = 0; i < 4; ++i) { v[i] = x[threadIdx.x + i * 256]; s += v[i]; }
    red[threadIdx.x] = s;
    __syncthreads();
    for (int st = 128; st; st >>= 1) {
        if (threadIdx.x < st) red[threadIdx.x] += red[threadIdx.x + st];
        __syncthreads();
    }
    float mean = red[0] * (1.0f / 1024.0f);
    __syncthreads();
    float s2 = 0.0f;
    #pragma unroll
    for (int i = 0; i < 4; ++i) { float d = v[i] - mean; s2 += d * d; }
    red[threadIdx.x] = s2;
    __syncthreads();
    for (int st = 128; st; st >>= 1) {
        if (threadIdx.x < st) red[threadIdx.x] += red[threadIdx.x + st];
        __syncthreads();
    }
    float rstd = rsqrtf(red[0] * (1.0f / 1024.0f) + 1e-5f);
    #pragma unroll
    for (int i = 0; i < 4; ++i) {
        int c = threadIdx.x + i * 256;
        float y = (v[i] - mean) * rstd * g[c] + b[c];
        if constexpr (GELU)
            y = 0.5f * y * (1.0f + erff(y * 0.70710678118654752440f));
        out[row * 1024 + c] = y;
    }
}

// ---------------------------------------------------------------------------
// rp[q,j] = sigmoid(dot(h[q], rg_w2[j]) + rg_b2[j]),  grid = 64*2 blocks
// ---------------------------------------------------------------------------
__global__ __launch_bounds__(256)
void rp_kernel(const float* __restrict__ h, const float* __restrict__ w2,
               const float* __restrict__ b2, float* __restrict__ rp)
{
    __shared__ float red[256];
    int q = blockIdx.x >> 1, j = blockIdx.x & 1;
    float s = 0.0f;
    for (int i = threadIdx.x; i < 1024; i += 256)
        s += h[(size_t)q * 1024 + i] * w2[(size_t)j * 1024 + i];
    red[threadIdx.x] = s;
    __syncthreads();
    for (int st = 128; st; st >>= 1) {
        if (threadIdx.x < st) red[threadIdx.x] += red[threadIdx.x + st];
        __syncthreads();
    }
    if (threadIdx.x == 0)
        rp[q * 2 + j] = 1.0f / (1.0f + expf(-(red[0] + b2[j])));
}

// ---------------------------------------------------------------------------
// Per-query: softmax over NP=4 (in-place on aw logits) + sample indices for
// the 3 levels.  grid=64 (q), block=64 (t = h*4 + p).
// ---------------------------------------------------------------------------
__global__ __launch_bounds__(64)
void samp_prep(float* __restrict__ aw, const float* __restrict__ offs,
               const float* __restrict__ rp, int* __restrict__ idx)
{
    __shared__ float lg[64];
    int q = blockIdx.x, t = threadIdx.x;
    lg[t] = aw[q * 64 + t];
    __syncthreads();
    int h = t >> 2, p = t & 3;
    float l0 = lg[h * 4 + 0], l1 = lg[h * 4 + 1], l2 = lg[h * 4 + 2], l3 = lg[h * 4 + 3];
    float m = fmaxf(fmaxf(l0, l1), fmaxf(l2, l3));
    float s = expf(l0 - m) + expf(l1 - m) + expf(l2 - m) + expf(l3 - m);
    aw[q * 64 + t] = expf(lg[t] - m) / s;

    float rpx = rp[q * 2 + 0], rpy = rp[q * 2 + 1];
    float ox = offs[q * 128 + h * 8 + p * 2 + 0];
    float oy = offs[q * 128 + h * 8 + p * 2 + 1];
    float spx = fminf(fmaxf(rpx + ox, 0.0f), 1.0f);
    float spy = fminf(fmaxf(rpy + oy, 0.0f), 1.0f);
    const int WH[3] = {24, 12, 8};
    #pragma unroll
    for (int l = 0; l < 3; ++l) {
        int Wsz = WH[l];
        int x0 = (int)floorf(spx * (float)(Wsz - 1));
        int y0 = (int)floorf(spy * (float)(Wsz - 1));
        idx[(l * 64 + q) * 64 + t] = y0 * Wsz + x0;
    }
}

// ---------------------------------------------------------------------------
// attn[b,q,h*64+d] = sum_l sum_p aw[q,h,p] * value[b, off_l+idx, h, d]
// grid = B*NQ = 4096, block = 256 (h = tid/16, d0 = (tid%16)*4)
// ---------------------------------------------------------------------------
__global__ __launch_bounds__(256)
void gather_attn(const float* __restrict__ value, const float* __restrict__ aw,
                 const int* __restrict__ idx, float* __restrict__ attn)
{
    __shared__ float s_aw[64];
    __shared__ int   s_idx[3 * 64];
    int bq = blockIdx.x;
    int b = bq >> 6, q = bq & 63;
    int t = threadIdx.x;
    if (t < 64)  s_aw[t] = aw[q * 64 + t];
    if (t < 192) s_idx[t] = idx[((t >> 6) * 64 + q) * 64 + (t & 63)];
    __syncthreads();

    int h = t >> 4;
    int d0 = (t & 15) << 2;
    v4f acc = {};
    const int lvl_off[3] = {0, 576, 720};
    #pragma unroll
    for (int l = 0; l < 3; ++l) {
        #pragma unroll
        for (int p = 0; p < 4; ++p) {
            float w = s_aw[h * 4 + p];
            int id = s_idx[l * 64 + h * 4 + p];
            const v4f v = *(const v4f*)(value +
                ((size_t)(b * 784 + lvl_off[l] + id)) * 1024 + h * 64 + d0);
            acc += w * v;
        }
    }
    *(v4f*)(attn + (size_t)bq * 1024 + h * 64 + d0) = acc;
}

// ---------------------------------------------------------------------------
extern "C" void kernel_launch(void* const* d_in, const int* in_sizes, int n_in,
                              void* d_out, int out_size, void* d_ws, size_t ws_size,
                              hipStream_t stream)
{
    (void)in_sizes; (void)n_in; (void)out_size; (void)ws_size;

    const float* features  = (const float*)d_in[0];
    const float* p1_w      = (const float*)d_in[1];
    const float* p1_b      = (const float*)d_in[2];
    const float* p2_w      = (const float*)d_in[3];
    const float* p2_b      = (const float*)d_in[4];
    const float* p3_w      = (const float*)d_in[5];
    const float* p3_b      = (const float*)d_in[6];
    const float* query_emb = (const float*)d_in[7];
    const float* rg_w1     = (const float*)d_in[8];
    const float* rg_b1     = (const float*)d_in[9];
    const float* rg_g      = (const float*)d_in[10];
    const float* rg_bb     = (const float*)d_in[11];
    const float* rg_w2     = (const float*)d_in[12];
    const float* rg_b2     = (const float*)d_in[13];
    const float* so_w      = (const float*)d_in[14];
    const float* so_b      = (const float*)d_in[15];
    const float* aw_w      = (const float*)d_in[16];
    const float* aw_b      = (const float*)d_in[17];
    const float* vp_w      = (const float*)d_in[18];
    const float* vp_b      = (const float*)d_in[19];
    const float* op_w      = (const float*)d_in[20];
    const float* op_b      = (const float*)d_in[21];
    const float* fln_g     = (const float*)d_in[22];
    const float* fln_b     = (const float*)d_in[23];
    const float* fin_w     = (const float*)d_in[24];
    const float* fin_b     = (const float*)d_in[25];
    float* out = (float*)d_out;

    float* ws = (float*)d_ws;
    size_t o = 0;
    auto alloc = [&](size_t n) { float* p = ws + o; o += (n + 63) & ~(size_t)63; return p; };

    float* Wf1   = alloc((size_t)1024 * 768);
    float* Wf2   = alloc((size_t)1024 * 3072);
    float* Wf3   = alloc((size_t)1024 * 6912);
    float* bf    = alloc(3 * 1024);
    float* value = alloc((size_t)64 * 784 * 1024);
    float* hq    = alloc(64 * 1024);
    float* hh    = alloc(64 * 1024);
    float* rp    = alloc(128);
    float* offs  = alloc(64 * 128);
    float* awbuf = alloc(64 * 64);
    int*   idxb  = (int*)alloc(3 * 64 * 64);
    float* attn  = alloc((size_t)4096 * 1024);
    float* t1    = alloc((size_t)4096 * 1024);

    // 1) weight fusion: Wf_l = vp_w @ p_l_w   (A@B mode)
    gemm_wmma<false, false><<<dim3(768 / 64, 1024 / 64), 256, 0, stream>>>(
        vp_w, p1_w, nullptr, Wf1, 1024, 768, 1024, 1024, 0, 0, 1, 1, 1);
    gemm_wmma<false, false><<<dim3(3072 / 64, 1024 / 64), 256, 0, stream>>>(
        vp_w, p2_w, nullptr, Wf2, 1024, 3072, 1024, 1024, 0, 0, 1, 1, 1);
    gemm_wmma<false, false><<<dim3(6912 / 64, 1024 / 64), 256, 0, stream>>>(
        vp_w, p3_w, nullptr, Wf3, 1024, 6912, 1024, 1024, 0, 0, 1, 1, 1);
    bias_fuse<<<dim3(1024, 3), 256, 0, stream>>>(vp_w, vp_b, p1_b, p2_b, p3_b, bf);

    // 2) value = concat levels, fused projection straight from features
    gemm_wmma<true, true><<<dim3(16, 576), 256, 0, stream>>>(
        features, Wf1, bf + 0,    value, 36864, 1024, 768,  576, 784, 0,   576, 24, 1);
    gemm_wmma<true, true><<<dim3(16, 144), 256, 0, stream>>>(
        features, Wf2, bf + 1024, value, 9216,  1024, 3072, 144, 784, 576, 144, 12, 2);
    gemm_wmma<true, true><<<dim3(16, 64), 256, 0, stream>>>(
        features, Wf3, bf + 2048, value, 4096,  1024, 6912, 64,  784, 720, 64,  8,  3);

    // 3) query-side path (batch-independent: 64 rows only)
    gemm_wmma<true, false><<<dim3(16, 1), 256, 0, stream>>>(
        query_emb, rg_w1, rg_b1, hq, 64, 1024, 1024, 64, 0, 0, 1, 1, 1);
    ln_kernel<true><<<64, 256, 0, stream>>>(hq, hh, rg_g, rg_bb);
    rp_kernel<<<128, 256, 0, stream>>>(hh, rg_w2, rg_b2, rp);
    gemm_wmma<true, false><<<dim3(2, 1), 256, 0, stream>>>(
        query_emb, so_w, so_b, offs, 64, 128, 1024, 64, 0, 0, 1, 1, 1);
    gemm_wmma<true, false><<<dim3(1, 1), 256, 0, stream>>>(
        query_emb, aw_w, aw_b, awbuf, 64, 64, 1024, 64, 0, 0, 1, 1, 1);
    samp_prep<<<64, 64, 0, stream>>>(awbuf, offs, rp, idxb);

    // 4) deformable gather
    gather_attn<<<4096, 256, 0, stream>>>(value, awbuf, idxb, attn);

    // 5) output projection -> LN -> final linear
    gemm_wmma<true, false><<<dim3(16, 64), 256, 0, stream>>>(
        attn, op_w, op_b, t1, 4096, 1024, 1024, 4096, 0, 0, 1, 1, 1);
    ln_kernel<false><<<4096, 256, 0, stream>>>(t1, t1, fln_g, fln_b);
    gemm_wmma<true, false><<<dim3(16, 64), 256, 0, stream>>>(
        t1, fin_w, fin_b, out, 4096, 1024, 1024, 4096, 0, 0, 1, 1, 1);
}